// GCN_13134009991660
// MI455X (gfx1250) — compile-verified
//
#include <hip/hip_runtime.h>
#include <hip/hip_bf16.h>

#define NODES 50000
#define EDGES 800000
#define DIM   96

typedef __attribute__((ext_vector_type(2))) float v2f;
typedef __attribute__((ext_vector_type(8))) float v8f;

// ---------------------------------------------------------------------------
// Zero-init workspace accumulator (graph-capture-safe, no hipMemset needed)
// ---------------------------------------------------------------------------
__global__ void gcn_zero_f32(float* __restrict__ p, int n) {
    int i = blockIdx.x * blockDim.x + threadIdx.x;
    if (i < n) p[i] = 0.0f;
}

// ---------------------------------------------------------------------------
// Edge aggregation: agg[dst] += x[src]  (PyG aggr='add')
// One wave32 per edge: lane l handles features l, l+32, l+64 -> fully
// coalesced 128B gather loads and coalesced global_atomic_add_f32 bursts.
// agg (19.2 MB) is L2-resident on MI455X (192 MB L2), so atomics stay on-chip.
// ---------------------------------------------------------------------------
__global__ void gcn_scatter_add(const float* __restrict__ x,
                                const int*   __restrict__ src,
                                const int*   __restrict__ dst,
                                float*       __restrict__ agg,
                                int nEdges) {
    int t = blockIdx.x * blockDim.x + threadIdx.x;
    int e = t >> 5;                 // edge id (one wave per edge)
    if (e >= nEdges) return;
    int lane = t & 31;
    int s = src[e];
    int d = dst[e];
    const float* xs = x   + (long long)s * DIM;
    float*       ad = agg + (long long)d * DIM;
#pragma unroll
    for (int i = 0; i < 3; ++i) {
        int f = lane + i * 32;      // 3 * 32 = 96 features
        atomicAdd(&ad[f], xs[f]);
    }
}

// ---------------------------------------------------------------------------
// Fused dual GEMM epilogue kernel:
//   out = [relu]( A1 @ W1  +  A2 @ W2  +  bias )
// One wave per 16x16 output tile; fp32 WMMA (V_WMMA_F32_16X16X4_F32) keeps
// full reference precision. K = 96 -> 24 WMMA steps per matmul (48 total).
//
// Fragment layouts per CDNA5 ISA 7.12.2 (wave32):
//   A 16x4 : lanes 0-15 rows M, VGPR0/1 = K0/K1 (lanes 0-15), K2/K3 (16-31)
//   B 4x16 : VGPR0 = row K0 (lanes 0-15) / K2 (lanes 16-31), VGPR1 = K1/K3
//   C 16x16: VGPR v -> M = v (lanes 0-15), M = v+8 (lanes 16-31), N = lane&15
// ---------------------------------------------------------------------------
__global__ void gcn_gemm_dual_wmma(const float* __restrict__ A1,
                                   const float* __restrict__ W1,
                                   const float* __restrict__ A2,
                                   const float* __restrict__ W2,
                                   const float* __restrict__ bias,
                                   float*       __restrict__ out,
                                   int nRowTiles, int doRelu) {
    const int colTiles = DIM / 16;                       // 6
    int wave = (blockIdx.x * blockDim.x + threadIdx.x) >> 5;
    if (wave >= nRowTiles * colTiles) return;            // uniform per wave

    int rowTile = wave / colTiles;
    int colTile = wave - rowTile * colTiles;
    int m0 = rowTile * 16;
    int n0 = colTile * 16;

    int lane = threadIdx.x & 31;
    int r    = lane & 15;       // row (A) / col (B,C) within tile
    int half = lane >> 4;       // half-wave select

    const float* a1row = A1 + (long long)(m0 + r) * DIM;
    const float* a2row = A2 + (long long)(m0 + r) * DIM;

    v8f c = {};

#pragma unroll
    for (int k0 = 0; k0 < DIM; k0 += 4) {
        int ka = k0 + half * 2;
        v2f a, b;
        a.x = a1row[ka];
        a.y = a1row[ka + 1];
        b.x = W1[ka * DIM + n0 + r];
        b.y = W1[(ka + 1) * DIM + n0 + r];
        c = __builtin_amdgcn_wmma_f32_16x16x4_f32(false, a, false, b,
                                                  (short)0, c, false, false);
    }
#pragma unroll
    for (int k0 = 0; k0 < DIM; k0 += 4) {
        int ka = k0 + half * 2;
        v2f a, b;
        a.x = a2row[ka];
        a.y = a2row[ka + 1];
        b.x = W2[ka * DIM + n0 + r];
        b.y = W2[(ka + 1) * DIM + n0 + r];
        c = __builtin_amdgcn_wmma_f32_16x16x4_f32(false, a, false, b,
                                                  (short)0, c, false, false);
    }

    float bv = bias[n0 + r];
#pragma unroll
    for (int v = 0; v < 8; ++v) {
        float val = c[v] + bv;
        if (doRelu) val = fmaxf(val, 0.0f);
        out[(long long)(m0 + v + half * 8) * DIM + n0 + r] = val;
    }
}

// ---------------------------------------------------------------------------
// Launch: zero -> scatter -> gemm(relu) -> zero -> scatter -> gemm
// ws layout: agg [N*D f32] | h [N*D f32]   (38.4 MB total)
// ---------------------------------------------------------------------------
extern "C" void kernel_launch(void* const* d_in, const int* in_sizes, int n_in,
                              void* d_out, int out_size, void* d_ws, size_t ws_size,
                              hipStream_t stream) {
    const float* x       = (const float*)d_in[0];
    const int*   ei      = (const int*)  d_in[1];   // [2, E]
    const float* W1_rel  = (const float*)d_in[2];
    const float* b1_rel  = (const float*)d_in[3];
    const float* W1_root = (const float*)d_in[4];
    const float* W2_rel  = (const float*)d_in[5];
    const float* b2_rel  = (const float*)d_in[6];
    const float* W2_root = (const float*)d_in[7];
    float*       out     = (float*)d_out;

    const int* src = ei;
    const int* dst = ei + EDGES;

    float* agg = (float*)d_ws;
    float* h   = agg + (size_t)NODES * DIM;

    const int nElems   = NODES * DIM;                 // 4.8M
    const int zeroBlks = (nElems + 255) / 256;
    const int scatBlks = (EDGES * 32 + 255) / 256;    // wave per edge
    const int nRowTiles = NODES / 16;                 // 3125 (exact)
    const int nTiles    = nRowTiles * (DIM / 16);     // 18750 waves
    const int gemmBlks  = (nTiles * 32 + 255) / 256;

    // ---- Layer 1: h = relu(agg(x) @ W1_rel + b1 + x @ W1_root) ----
    gcn_zero_f32<<<zeroBlks, 256, 0, stream>>>(agg, nElems);
    gcn_scatter_add<<<scatBlks, 256, 0, stream>>>(x, src, dst, agg, EDGES);
    gcn_gemm_dual_wmma<<<gemmBlks, 256, 0, stream>>>(agg, W1_rel, x, W1_root,
                                                     b1_rel, h, nRowTiles, 1);

    // ---- Layer 2: out = agg(h) @ W2_rel + b2 + h @ W2_root ----
    gcn_zero_f32<<<zeroBlks, 256, 0, stream>>>(agg, nElems);
    gcn_scatter_add<<<scatBlks, 256, 0, stream>>>(h, src, dst, agg, EDGES);
    gcn_gemm_dual_wmma<<<gemmBlks, 256, 0, stream>>>(agg, W2_rel, h, W2_root,
                                                     b2_rel, out, nRowTiles, 0);
}